// Attender_68221260529728
// MI455X (gfx1250) — compile-verified
//
#include <hip/hip_runtime.h>

// CDNA5 / gfx1250, wave32. fp32 end-to-end using V_WMMA_F32_16X16X4_F32,
// with gfx1250 async global->LDS staging (ASYNCcnt) and double-buffered tiles.
typedef __attribute__((ext_vector_type(2))) float v2f;
typedef __attribute__((ext_vector_type(8))) float v8f;

#define BATCH   8
#define SEQ     2048
#define DIM     1024
#define WIN     20
#define NEG_BIG (-1e8f)

// Kernel 1 tiling: 128(M) x 128(N) per 256-thread block, K staged 32 wide.
#define KC1     32
#define LDA1    36                    // padded row stride (floats): 144 B, 16B-aligned, conflict-free
#define TILE1   (128 * LDA1)          // one A or B buffer (floats)
#define SMEM1_BYTES (4 * TILE1 * 4)   // {A0,B0,A1,B1} = 73,728 B (dynamic LDS; 320KB/WGP on gfx1250)
// Kernel 2 staging: 16 x 512 K-chunks.
#define KC2     512
#define LDC2    516                   // 2064 B rows, 16B-aligned, conflict-free
#define LDS_S   68                    // padded stride for the 16x64 sims/probs tile

static __device__ __forceinline__ v8f wmma4(v2f a, v2f b, v8f c) {
  // D = A(16x4 f32) * B(4x16 f32) + C(16x16 f32)
  return __builtin_amdgcn_wmma_f32_16x16x4_f32(false, a, false, b, (short)0, c, false, false);
}
static __device__ __forceinline__ v8f zero8() {
  v8f z = {0.f, 0.f, 0.f, 0.f, 0.f, 0.f, 0.f, 0.f};
  return z;
}

// gfx1250 async copy: 16B global -> LDS, tracked by ASYNCcnt (no VGPR round-trip).
// LDS offset = low 32 bits of the generic pointer (flat aperture keeps the DS
// byte offset in addr[31:0]).
static __device__ __forceinline__ void async_copy_b128(const float* __restrict__ gsrc,
                                                       const float* ldst) {
  unsigned           loff = (unsigned)(unsigned long long)ldst;
  unsigned long long ga   = (unsigned long long)gsrc;
  asm volatile("global_load_async_to_lds_b128 %0, %1, off"
               :
               : "v"(loff), "v"(ga)
               : "memory");
}
static __device__ __forceinline__ void wait_async() {
  asm volatile("s_wait_asynccnt 0x0" ::: "memory");
}

// ---------------------------------------------------------------------------
// Kernel 1: keys_r[m, n] = sum_k keys[m, k] * W[n, k] + bias[n]
//   M = BATCH*SEQ = 16384, N = DIM, K = DIM.
//   Block = 8 waves (256 thr), tile 128(M) x 128(N); A and B double-buffered in
//   dynamic LDS via async b128 loads; each wave computes 16x128 (8 v8f accs).
// ---------------------------------------------------------------------------
__global__ __launch_bounds__(256, 1)
void reduce_gemm_kernel(const float* __restrict__ keys, const float* __restrict__ W,
                        const float* __restrict__ bias, float* __restrict__ keys_r) {
  extern __shared__ float smem[];
  float* sA0 = smem;
  float* sB0 = smem + TILE1;
  float* sA1 = smem + 2 * TILE1;
  float* sB1 = smem + 3 * TILE1;

  const int m0   = blockIdx.x * 128;
  const int n0   = blockIdx.y * 128;
  const int tid  = threadIdx.x;
  const int wave = tid >> 5;
  const int lane = tid & 31;
  const int half = lane >> 4;   // 0: K pair {0,1}, 1: K pair {2,3}
  const int l16  = lane & 15;

  v8f acc[8];
#pragma unroll
  for (int s = 0; s < 8; ++s) acc[s] = zero8();

  // Issue a 128x32 A-tile + 128x32 B-tile stage (async, b128 granules).
  auto stage = [&](int kc, float* sA, float* sB) {
#pragma unroll
    for (int t = 0; t < 4; ++t) {
      const int i   = tid + t * 256;   // 0..1023
      const int row = i >> 3;
      const int c4  = i & 7;
      async_copy_b128(keys + (size_t)(m0 + row) * DIM + kc + c4 * 4,
                      sA + row * LDA1 + c4 * 4);
      async_copy_b128(W + (size_t)(n0 + row) * DIM + kc + c4 * 4,
                      sB + row * LDA1 + c4 * 4);
    }
  };

  // One 32-wide K chunk of WMMAs. B operands gathered into distinct registers
  // first so ds_load latency overlaps the WMMA burst (avoids s_wait_dscnt 0).
  auto compute = [&](const float* sA, const float* sB) {
    const float* aLane = sA + (wave * 16 + l16) * LDA1 + half * 2;
    const float* bLane = sB + l16 * LDA1 + half * 2;
#pragma unroll
    for (int k = 0; k < KC1; k += 4) {
      v2f a = *(const v2f*)(aLane + k);
      v2f b[8];
#pragma unroll
      for (int s = 0; s < 8; ++s)
        b[s] = *(const v2f*)(bLane + s * 16 * LDA1 + k);
#pragma unroll
      for (int s = 0; s < 8; ++s)
        acc[s] = wmma4(a, b[s], acc[s]);
    }
  };

  // Prologue: chunk 0 -> buffer 0.
  stage(0, sA0, sB0);
  wait_async();
  __syncthreads();

  // Ping-pong: stage chunk i+1 while computing chunk i.
  for (int kc = 0; kc < DIM; kc += 2 * KC1) {
    stage(kc + KC1, sA1, sB1);          // overlap with compute on buf0
    compute(sA0, sB0);
    wait_async();
    __syncthreads();

    if (kc + 2 * KC1 < DIM)
      stage(kc + 2 * KC1, sA0, sB0);    // overlap with compute on buf1
    compute(sA1, sB1);
    wait_async();
    __syncthreads();
  }

#pragma unroll
  for (int s = 0; s < 8; ++s) {
    const int   col = n0 + s * 16 + l16;
    const float bv  = bias[col];
#pragma unroll
    for (int r = 0; r < 8; ++r)
      keys_r[(size_t)(m0 + wave * 16 + r + half * 8) * DIM + col] = acc[s][r] + bv;
  }
}

// ---------------------------------------------------------------------------
// Kernel 2: banded attention for one (batch, 16-query tile). 128 threads.
//   Window: 64 keys starting at q0-24 (covers |q-k|<=20 for all 16 queries).
//   Phase 1: sims(16x64) = Q @ keys_r^T via WMMA (Q async-staged in LDS).
//   Softmax per row (masked entries -> exp underflows to exact 0, like the ref).
//   Phase 2: out(16x1024) = P(16x64) @ V(64x1024), V = original keys rows.
// ---------------------------------------------------------------------------
__global__ __launch_bounds__(128, 1)
void band_attn_kernel(const float* __restrict__ Q, const float* __restrict__ keys,
                      const float* __restrict__ keys_r, float* __restrict__ out) {
  __shared__ float sQ[16 * LDC2];
  __shared__ float sS[16 * LDS_S];

  const int b      = blockIdx.y;
  const int q0     = blockIdx.x * 16;
  const int kstart = q0 - 24;
  const int tid    = threadIdx.x;
  const int wave   = tid >> 5;
  const int lane   = tid & 31;
  const int half   = lane >> 4;
  const int l16    = lane & 15;

  // ---- Phase 1: sims = Q @ keys_r^T over the 64-wide window ----
  const int    kcol   = kstart + wave * 16 + l16;          // key index of this lane's B column
  const int    kclamp = min(max(kcol, 0), SEQ - 1);        // clamp addr; value masked later
  const float* krow   = keys_r + ((size_t)b * SEQ + kclamp) * DIM + half * 2;

  v8f sacc = zero8();
  for (int kc = 0; kc < DIM; kc += KC2) {
    __syncthreads();
    // Stage Q[16][kc..kc+511]: 2048 float4, 16 per thread, async to LDS.
#pragma unroll
    for (int t = 0; t < 16; ++t) {
      const int i   = tid + t * 128;   // 0..2047
      const int row = i >> 7;
      const int c4  = i & 127;
      async_copy_b128(Q + ((size_t)b * SEQ + q0 + row) * DIM + kc + c4 * 4,
                      sQ + row * LDC2 + c4 * 4);
    }
    wait_async();
    __syncthreads();

    const float* aLane = sQ + l16 * LDC2 + half * 2;
#pragma unroll 2
    for (int k = 0; k < KC2; k += 16) {
      v2f a[4], bb[4];
#pragma unroll
      for (int j = 0; j < 4; ++j) {
        a[j]  = *(const v2f*)(aLane + k + j * 4);
        bb[j] = *(const v2f*)(krow + kc + k + j * 4);
      }
#pragma unroll
      for (int j = 0; j < 4; ++j)
        sacc = wmma4(a[j], bb[j], sacc);
    }
  }

  // Apply band + bounds mask, park sims in LDS.
#pragma unroll
  for (int r = 0; r < 8; ++r) {
    const int q = q0 + r + half * 8;
    const int d = q - kcol;
    float v = sacc[r];
    if (kcol < 0 || kcol >= SEQ || d > WIN || d < -WIN) v = NEG_BIG;
    sS[(r + half * 8) * LDS_S + wave * 16 + l16] = v;
  }
  __syncthreads();

  // Row softmax over 64 window entries (one thread per query row; 16 rows).
  if (tid < 16) {
    float m = -3.0e38f;
    for (int j = 0; j < 64; ++j) m = fmaxf(m, sS[tid * LDS_S + j]);
    float sum = 0.f;
    for (int j = 0; j < 64; ++j) {
      const float e = __expf(sS[tid * LDS_S + j] - m);  // masked -> exactly 0
      sS[tid * LDS_S + j] = e;
      sum += e;
    }
    const float inv = 1.f / sum;
    for (int j = 0; j < 64; ++j) sS[tid * LDS_S + j] *= inv;
  }
  __syncthreads();

  // ---- Phase 2: out = P @ V. A-operand (P) from LDS, held in registers. ----
  v2f aP[16];
#pragma unroll
  for (int kk = 0; kk < 16; ++kk)
    aP[kk] = *(const v2f*)(sS + l16 * LDS_S + kk * 4 + half * 2);

  const float* vbase = keys + (size_t)b * SEQ * DIM;
  for (int nt = 0; nt < 16; ++nt) {
    const int n = wave * 256 + nt * 16 + l16;
    v8f acc = zero8();
#pragma unroll
    for (int kk = 0; kk < 16; ++kk) {
      const int kr0 = kstart + kk * 4 + half * 2;
      const int c0  = min(max(kr0, 0), SEQ - 1);       // OOB rows multiply P==0
      const int c1  = min(max(kr0 + 1, 0), SEQ - 1);
      v2f bb;
      bb.x = vbase[(size_t)c0 * DIM + n];
      bb.y = vbase[(size_t)c1 * DIM + n];
      acc = wmma4(aP[kk], bb, acc);
    }
#pragma unroll
    for (int r = 0; r < 8; ++r)
      out[((size_t)b * SEQ + q0 + r + half * 8) * DIM + n] = acc[r];
  }
}

// ---------------------------------------------------------------------------
extern "C" void kernel_launch(void* const* d_in, const int* in_sizes, int n_in,
                              void* d_out, int out_size, void* d_ws, size_t ws_size,
                              hipStream_t stream) {
  const float* queries = (const float*)d_in[0];  // [8, 2048, 1024]
  const float* keys    = (const float*)d_in[1];  // [8, 2048, 1024]
  const float* W       = (const float*)d_in[2];  // [1024, 1024] (N-major by K)
  const float* bias    = (const float*)d_in[3];  // [1024]
  float* out    = (float*)d_out;
  float* keys_r = (float*)d_ws;                  // needs BATCH*SEQ*DIM*4 = 64 MiB

  dim3 g1((BATCH * SEQ) / 128, DIM / 128);       // 128 x 8 blocks, 256 threads
  reduce_gemm_kernel<<<g1, 256, SMEM1_BYTES, stream>>>(keys, W, bias, keys_r);

  dim3 g2(SEQ / 16, BATCH);                      // 128 x 8 blocks, 128 threads
  band_attn_kernel<<<g2, 128, 0, stream>>>(queries, keys, keys_r, out);
}